// ModulatedLinear_39444979646820
// MI455X (gfx1250) — compile-verified
//
#include <hip/hip_runtime.h>

#define IN_F   1024
#define OUT_F  1024
#define BATCH  32
#define SEQ    1024
#define EPS    1e-8f

#define TILE_M 128
#define TILE_N 128
#define TILE_K 16
#define NCHUNK (IN_F / TILE_K)
#define APAD   20    // A row stride (floats): 80B, 16B-aligned, conflict-free reads
#define BPAIRS (TILE_K / 2)
#define BPITCH 288   // dwords per B pair-row: 288 % 64 == 32 -> half-waves in disjoint bank windows

typedef __attribute__((ext_vector_type(2))) float v2f;
typedef __attribute__((ext_vector_type(8))) float v8f;

// CDNA5 async global->LDS copy (ASYNCcnt-tracked), GVS addressing:
//   lds[lds_addr] = mem[saddr + vaddr], 16 bytes per lane.
__device__ __forceinline__ void async_copy_b128(unsigned lds_addr,
                                                const float* __restrict__ sbase,
                                                int vbyte_off) {
    asm volatile("global_load_async_to_lds_b128 %0, %1, %2"
                 :: "v"(lds_addr), "v"(vbyte_off), "s"(sbase)
                 : "memory");
}
__device__ __forceinline__ void wait_async0() {
    asm volatile("s_wait_asynccnt 0x0" ::: "memory");
}

// ---------------------------------------------------------------------------
// Kernel 1: d[b][o] = rsqrt( (1/IN_F) * sum_i y[b][i]^2 * W[i][o]^2 + eps )
// ---------------------------------------------------------------------------
__global__ __launch_bounds__(256)
void demod_kernel(const float* __restrict__ y,
                  const float* __restrict__ W,
                  float* __restrict__ d) {
    const int b = blockIdx.y;
    const int o = blockIdx.x * 256 + threadIdx.x;
    const float* yrow = y + b * IN_F;
    float acc = 0.0f;
    for (int i = 0; i < IN_F; ++i) {
        const float yv = yrow[i];                 // uniform -> scalar load
        const float wv = W[i * OUT_F + o];        // coalesced
        acc += (yv * yv) * (wv * wv);
    }
    acc *= (1.0f / (float)IN_F);                  // scale^2
    d[b * OUT_F + o] = rsqrtf(acc + EPS);
}

// ---------------------------------------------------------------------------
// Kernel 2: out[b][s][o] = ( Efou[b] @ (diag(ys[b]) * W) )[s][o] * d[b][o]
// 128x128 block tile, 8 waves x (64x32), f32 WMMA, double-buffered LDS.
// A (Efou) streamed via async-to-LDS; B (W) K-pair interleaved in LDS so
// fragments are single even-aligned ds_load_b64s.
// ---------------------------------------------------------------------------
__global__ __launch_bounds__(256)
void modlinear_gemm(const float* __restrict__ Efou,
                    const float* __restrict__ y,
                    const float* __restrict__ W,
                    const float* __restrict__ d,
                    float* __restrict__ out) {
    __shared__ __align__(16) float As[2][TILE_M * APAD];     // raw Efou tiles
    __shared__ __align__(16) float Bs[2][BPAIRS * BPITCH];   // ys*W, K-pair interleaved
    __shared__ float ysh[IN_F];                              // ys = y[b] * scale

    const int b  = blockIdx.z;
    const int m0 = blockIdx.y * TILE_M;
    const int n0 = blockIdx.x * TILE_N;

    const int tid  = threadIdx.x;
    const int wave = tid >> 5;
    const int lane = tid & 31;
    const int hl   = lane & 15;
    const int hi   = lane >> 4;

    const int wm = (wave & 1) * 64;
    const int wn = (wave >> 1) * 32;

    const float* Ab = Efou + (size_t)b * SEQ * IN_F;

    // ---- ys into LDS ----
    const float scale = 0.03125f;  // 1/sqrt(1024)
    for (int i = tid; i < IN_F; i += 256)
        ysh[i] = y[b * IN_F + i] * scale;

    // A async-fill mapping: 512 float4 per chunk, 2 per thread.
    const unsigned asBase[2] = {(unsigned)(size_t)&As[0][0],
                                (unsigned)(size_t)&As[1][0]};
    const int aRow0 = tid >> 2,         aCf0 = (tid & 3) * 4;
    const int aRow1 = (tid + 256) >> 2, aCf1 = (tid & 3) * 4;
    const unsigned aLds0 = (unsigned)(aRow0 * APAD + aCf0) * 4u;
    const unsigned aLds1 = (unsigned)(aRow1 * APAD + aCf1) * 4u;
    const int aOff0 = ((m0 + aRow0) * IN_F + aCf0) * 4;   // byte offset (+ k0*4)
    const int aOff1 = ((m0 + aRow1) * IN_F + aCf1) * 4;

    // B fill mapping: 8 pair-rows x 64 col-pairs; 2 passes of 4 pair-rows.
    const int pr = tid >> 6;           // 0..3 (+4 on pass 1)
    const int cp = tid & 63;           // column-pair index (col = 2*cp)

    v8f acc[4][2];
#pragma unroll
    for (int mt = 0; mt < 4; ++mt)
#pragma unroll
        for (int nt = 0; nt < 2; ++nt)
            acc[mt][nt] = (v8f){0.f, 0.f, 0.f, 0.f, 0.f, 0.f, 0.f, 0.f};

    __syncthreads();  // ysh visible

    // ---- prologue: fill buffer 0 (chunk 0) ----
    async_copy_b128(asBase[0] + aLds0, Ab, aOff0);
    async_copy_b128(asBase[0] + aLds1, Ab, aOff1);
#pragma unroll
    for (int p = 0; p < 2; ++p) {
        const int prr = pr + p * 4;
        const int k   = 2 * prr;
        const float2 w0 = *(const float2*)(W + (size_t)(k + 0) * OUT_F + n0 + 2 * cp);
        const float2 w1 = *(const float2*)(W + (size_t)(k + 1) * OUT_F + n0 + 2 * cp);
        const float ym0 = ysh[k], ym1 = ysh[k + 1];
        float4 m; m.x = w0.x * ym0; m.y = w1.x * ym1; m.z = w0.y * ym0; m.w = w1.y * ym1;
        *(float4*)(&Bs[0][prr * BPITCH + 4 * cp]) = m;
    }
    wait_async0();
    __syncthreads();

    for (int kc = 0; kc < NCHUNK; ++kc) {
        const int cur = kc & 1;
        const int nxt = cur ^ 1;
        const int k1  = (kc + 1) * TILE_K;      // next chunk K base (floats)
        const bool pf = (kc + 1) < NCHUNK;

        // ---- issue next-chunk fills: async A + staged B global loads ----
        float2 bstage[2][2];
        if (pf) {
            async_copy_b128(asBase[nxt] + aLds0, Ab, aOff0 + k1 * 4);
            async_copy_b128(asBase[nxt] + aLds1, Ab, aOff1 + k1 * 4);
#pragma unroll
            for (int p = 0; p < 2; ++p) {
                const int k = k1 + 2 * (pr + p * 4);
                bstage[p][0] = *(const float2*)(W + (size_t)(k + 0) * OUT_F + n0 + 2 * cp);
                bstage[p][1] = *(const float2*)(W + (size_t)(k + 1) * OUT_F + n0 + 2 * cp);
            }
        }

        // ---- compute on current buffer: 4 K-steps of 4 -> 32 WMMAs ----
        const float* Ac = &As[cur][0];
        const float* Bc = &Bs[cur][0];
#pragma unroll
        for (int kk = 0; kk < 4; ++kk) {
            const int kb = kk * 4 + hi * 2;     // lanes 0-15: K=kb,kb+1; 16-31: +2

            v2f af[4];
#pragma unroll
            for (int mt = 0; mt < 4; ++mt) {
                const float* ap = &Ac[(wm + mt * 16 + hl) * APAD + kb];
                af[mt].x = ap[0];               // adjacent -> ds_load_b64
                af[mt].y = ap[1];
            }
            v2f bf[2];
#pragma unroll
            for (int nt = 0; nt < 2; ++nt) {
                const int col = wn + nt * 16 + hl;
                const float* bp = &Bc[(kk * 2 + hi) * BPITCH + 2 * col];
                bf[nt].x = bp[0];               // K-pair interleave -> ds_load_b64
                bf[nt].y = bp[1];
            }
#pragma unroll
            for (int mt = 0; mt < 4; ++mt)
#pragma unroll
                for (int nt = 0; nt < 2; ++nt)
                    acc[mt][nt] = __builtin_amdgcn_wmma_f32_16x16x4_f32(
                        false, af[mt], false, bf[nt],
                        (short)0, acc[mt][nt], false, false);
        }

        // ---- commit staged B (modulated, interleaved) into next buffer ----
        if (pf) {
#pragma unroll
            for (int p = 0; p < 2; ++p) {
                const int prr = pr + p * 4;
                const int k   = k1 + 2 * prr;
                const float ym0 = ysh[k], ym1 = ysh[k + 1];
                float4 m;
                m.x = bstage[p][0].x * ym0; m.y = bstage[p][1].x * ym1;
                m.z = bstage[p][0].y * ym0; m.w = bstage[p][1].y * ym1;
                *(float4*)(&Bs[nxt][prr * BPITCH + 4 * cp]) = m;
            }
        }

        wait_async0();       // own async fills of As[nxt] done
        __syncthreads();     // everyone done reading cur / writing nxt
    }

    // ---- epilogue: scale by d[b][n], store ----
    float* ob = out + (size_t)b * SEQ * OUT_F;
    const float* db = d + b * OUT_F;
#pragma unroll
    for (int nt = 0; nt < 2; ++nt) {
        const int ncol = n0 + wn + nt * 16 + hl;
        const float dv = db[ncol];
#pragma unroll
        for (int mt = 0; mt < 4; ++mt) {
            const int mbase = m0 + wm + mt * 16 + hi * 8;   // C/D row layout
#pragma unroll
            for (int r = 0; r < 8; ++r) {
                ob[(size_t)(mbase + r) * OUT_F + ncol] = acc[mt][nt][r] * dv;
            }
        }
    }
}

// ---------------------------------------------------------------------------
extern "C" void kernel_launch(void* const* d_in, const int* in_sizes, int n_in,
                              void* d_out, int out_size, void* d_ws, size_t ws_size,
                              hipStream_t stream) {
    (void)in_sizes; (void)n_in; (void)out_size; (void)ws_size;

    const float* Efou = (const float*)d_in[0];   // (32,1024,1024)
    const float* y    = (const float*)d_in[1];   // (32,1024)
    const float* W    = (const float*)d_in[2];   // (1024,1024)
    float* out        = (float*)d_out;           // (32,1024,1024)
    float* dbuf       = (float*)d_ws;            // (32,1024)

    dim3 gA(OUT_F / 256, BATCH);
    demod_kernel<<<gA, 256, 0, stream>>>(y, W, dbuf);

    dim3 gB(OUT_F / TILE_N, SEQ / TILE_M, BATCH);
    modlinear_gemm<<<gB, 256, 0, stream>>>(Efou, y, W, dbuf, out);
}